// Affinity_5583457485027
// MI455X (gfx1250) — compile-verified
//
#include <hip/hip_runtime.h>
#include <hip/hip_bf16.h>

typedef __attribute__((ext_vector_type(2))) float v2f;
typedef __attribute__((ext_vector_type(8))) float v8f;

#define NN 72            // nodes per graph
#define NE 288           // edges per graph
#define DD 64            // feature dim
#define NM (NN * NN)     // 5184 = output rows/cols

// workspace layout (in floats)
#define AR_OFF   0                        // relu(lamda1)  [64,64]
#define BR_OFF   (AR_OFF + DD * DD)      // relu(lamda2)  [64,64]
#define XSUM_OFF (BR_OFF + DD * DD)      // F1[src1]+F1[dst1]  [288,64]
#define YS_OFF   (XSUM_OFF + NE * DD)    // F2[src2]      [288,64]
#define YD_OFF   (YS_OFF + NE * DD)      // F2[dst2]      [288,64]
#define Z_OFF    (YD_OFF + NE * DD)      // Ys*Ar^T+Yd*Br^T [288,64]
#define ME_OFF   (Z_OFF + NE * DD)       // edge affinity [288,288]
#define MP_OFF   (ME_OFF + NE * NE)      // node affinity [72,72]

// ---------------------------------------------------------------------------
// One 16x16 f32 tile of C += L * R^T  (both L,R row-major, leading dim = ld).
// Uses V_WMMA_F32_16X16X4_F32; full wave32, EXEC all ones.
// Lane layout: A: (m=lane&15, k0=k+2*(lane>>4)), B: (n=lane&15, same k),
// D: row = v + 8*(lane>>4), col = lane&15.
// ---------------------------------------------------------------------------
__device__ __forceinline__ v8f wmma_nt_accum(const float* __restrict__ L, int ldl, int mr,
                                             const float* __restrict__ R, int ldr, int nr,
                                             int K, int half, v8f acc) {
#pragma unroll
    for (int k = 0; k < 64; k += 4) {   // K is always 64 here
        int k0 = k + 2 * half;
        v2f a = *(const v2f*)(L + mr * ldl + k0);
        v2f b = *(const v2f*)(R + nr * ldr + k0);
        acc = __builtin_amdgcn_wmma_f32_16x16x4_f32(
            /*neg_a=*/false, a, /*neg_b=*/false, b,
            /*c_mod=*/(short)0, acc, /*reuse_a=*/false, /*reuse_b=*/false);
    }
    (void)K;
    return acc;
}

// ---------------------------------------------------------------------------
// Prep: relu(lamda1/2), Xsum = F1[src1]+F1[dst1], gather Ys=F2[src2], Yd=F2[dst2]
// ---------------------------------------------------------------------------
__global__ void prep_kernel(const float* __restrict__ F1, const float* __restrict__ F2,
                            const float* __restrict__ l1, const float* __restrict__ l2,
                            const int* __restrict__ src1, const int* __restrict__ dst1,
                            const int* __restrict__ src2, const int* __restrict__ dst2,
                            float* __restrict__ ws) {
    int idx = blockIdx.x * blockDim.x + threadIdx.x;
    const int LAM = DD * DD;          // 4096
    const int GAT = NE * DD;          // 18432
    if (idx < LAM) {
        float v = l1[idx];
        ws[AR_OFF + idx] = v > 0.f ? v : 0.f;
    } else if (idx < 2 * LAM) {
        int t = idx - LAM;
        float v = l2[t];
        ws[BR_OFF + t] = v > 0.f ? v : 0.f;
    } else if (idx < 2 * LAM + GAT) {
        int t = idx - 2 * LAM;
        int e = t >> 6, k = t & 63;
        ws[XSUM_OFF + t] = F1[src1[e] * DD + k] + F1[dst1[e] * DD + k];
    } else if (idx < 2 * LAM + 2 * GAT) {
        int t = idx - (2 * LAM + GAT);
        int e = t >> 6, k = t & 63;
        ws[YS_OFF + t] = F2[src2[e] * DD + k];
    } else if (idx < 2 * LAM + 3 * GAT) {
        int t = idx - (2 * LAM + 2 * GAT);
        int e = t >> 6, k = t & 63;
        ws[YD_OFF + t] = F2[dst2[e] * DD + k];
    }
}

// ---------------------------------------------------------------------------
// Z[288,64] = Ys * Ar^T + Yd * Br^T      (18 x 4 tiles, one wave each)
// ---------------------------------------------------------------------------
__global__ void z_gemm_kernel(const float* __restrict__ ws_in, float* __restrict__ ws) {
    const float* ys = ws_in + YS_OFF;
    const float* yd = ws_in + YD_OFF;
    const float* ar = ws_in + AR_OFF;
    const float* br = ws_in + BR_OFF;
    float* z = ws + Z_OFF;

    int tile = blockIdx.x;            // 0..71
    int tm = tile >> 2, tn = tile & 3;
    int lane = threadIdx.x & 31;
    int half = lane >> 4;
    int mr = tm * 16 + (lane & 15);
    int nr = tn * 16 + (lane & 15);

    v8f acc = {};
    acc = wmma_nt_accum(ys, DD, mr, ar, DD, nr, DD, half, acc);
    acc = wmma_nt_accum(yd, DD, mr, br, DD, nr, DD, half, acc);

    int col = tn * 16 + (lane & 15);
#pragma unroll
    for (int v = 0; v < 8; ++v)
        z[(tm * 16 + v + 8 * half) * DD + col] = acc[v];
}

// ---------------------------------------------------------------------------
// Me[288,288] = Xsum * Z^T              (18 x 18 tiles)
// ---------------------------------------------------------------------------
__global__ void me_gemm_kernel(const float* __restrict__ ws_in, float* __restrict__ ws) {
    const float* xs = ws_in + XSUM_OFF;
    const float* z  = ws_in + Z_OFF;
    float* me = ws + ME_OFF;

    int tm = blockIdx.y, tn = blockIdx.x;
    int lane = threadIdx.x & 31;
    int half = lane >> 4;
    int mr = tm * 16 + (lane & 15);
    int nr = tn * 16 + (lane & 15);

    v8f acc = {};
    acc = wmma_nt_accum(xs, DD, mr, z, DD, nr, DD, half, acc);

    int col = tn * 16 + (lane & 15);
#pragma unroll
    for (int v = 0; v < 8; ++v)
        me[(tm * 16 + v + 8 * half) * NE + col] = acc[v];
}

// ---------------------------------------------------------------------------
// Mp[72,72] = U1 * U2^T                 (5 x 5 tiles, clamped loads, guarded store)
// ---------------------------------------------------------------------------
__global__ void mp_gemm_kernel(const float* __restrict__ U1, const float* __restrict__ U2,
                               float* __restrict__ ws) {
    float* mp = ws + MP_OFF;
    int tm = blockIdx.y, tn = blockIdx.x;
    int lane = threadIdx.x & 31;
    int half = lane >> 4;
    int mr = tm * 16 + (lane & 15);
    int nr = tn * 16 + (lane & 15);
    int mrc = mr < NN ? mr : NN - 1;   // clamp: garbage rows/cols never stored
    int nrc = nr < NN ? nr : NN - 1;

    v8f acc = {};
    acc = wmma_nt_accum(U1, DD, mrc, U2, DD, nrc, DD, half, acc);

    int col = tn * 16 + (lane & 15);
#pragma unroll
    for (int v = 0; v < 8; ++v) {
        int row = tm * 16 + v + 8 * half;
        if (row < NN && col < NN) mp[row * NN + col] = acc[v];
    }
}

// ---------------------------------------------------------------------------
// Zero-fill the 107.5 MB output with b128 stores (bandwidth-bound floor)
// ---------------------------------------------------------------------------
__global__ void zero_out_kernel(float4* __restrict__ out, int n4) {
    int stride = gridDim.x * blockDim.x;
    for (int i = blockIdx.x * blockDim.x + threadIdx.x; i < n4; i += stride)
        out[i] = make_float4(0.f, 0.f, 0.f, 0.f);
}

// ---------------------------------------------------------------------------
// Scatter: each edge pair (j in E2, i in E1) adds Me[j,i] to 16 positions.
// 4 threads per pair (one (a,c) combo each), 4 atomics per thread.
// Output fits in 192MB L2 -> atomics stay on-chip.
// ---------------------------------------------------------------------------
__global__ void scatter_kernel(const float* __restrict__ ws,
                               const int* __restrict__ src1, const int* __restrict__ dst1,
                               const int* __restrict__ src2, const int* __restrict__ dst2,
                               float* __restrict__ out) {
    const float* me = ws + ME_OFF;
    int tid = blockIdx.x * blockDim.x + threadIdx.x;
    if (tid >= NE * NE * 4) return;
    int pair = tid >> 2;
    int q = tid & 3;
    int j = pair / NE;                 // graph-2 edge -> (a,c)
    int i = pair - j * NE;             // graph-1 edge -> (b,d)
    float val = me[j * NE + i];        // W[j,i] == Me[j,i] since e1==e2
    int s2 = src2[j], t2 = dst2[j];
    int s1 = src1[i], t1 = dst1[i];
    int a = (q & 1) ? t2 : s2;
    int c = (q & 2) ? t2 : s2;
    int cb = c * NN;
    atomicAdd(&out[(a * NN + s1) * NM + cb + s1], val);
    atomicAdd(&out[(a * NN + s1) * NM + cb + t1], val);
    atomicAdd(&out[(a * NN + t1) * NM + cb + s1], val);
    atomicAdd(&out[(a * NN + t1) * NM + cb + t1], val);
}

// ---------------------------------------------------------------------------
// Diagonal add of Mp (runs after scatter; stream-ordered so plain RMW is safe)
// ---------------------------------------------------------------------------
__global__ void diag_kernel(const float* __restrict__ ws, float* __restrict__ out) {
    const float* mp = ws + MP_OFF;
    int idx = blockIdx.x * blockDim.x + threadIdx.x;
    if (idx < NM) out[idx * NM + idx] += mp[idx];
}

extern "C" void kernel_launch(void* const* d_in, const int* in_sizes, int n_in,
                              void* d_out, int out_size, void* d_ws, size_t ws_size,
                              hipStream_t stream) {
    const float* F1 = (const float*)d_in[0];
    const float* F2 = (const float*)d_in[1];
    const float* U1 = (const float*)d_in[2];
    const float* U2 = (const float*)d_in[3];
    const float* l1 = (const float*)d_in[4];
    const float* l2 = (const float*)d_in[5];
    const int* src1 = (const int*)d_in[6];
    const int* dst1 = (const int*)d_in[7];
    const int* src2 = (const int*)d_in[8];
    const int* dst2 = (const int*)d_in[9];
    float* out = (float*)d_out;
    float* ws = (float*)d_ws;

    // 1) zero-fill output (dominant memory cost: 107.5 MB)
    int n4 = (NM * NM) / 4;   // 6,718,464 float4
    zero_out_kernel<<<4096, 256, 0, stream>>>((float4*)out, n4);

    // 2) prep: relu(lamda), Xsum, gathers   (63488 elements)
    prep_kernel<<<248, 256, 0, stream>>>(F1, F2, l1, l2, src1, dst1, src2, dst2, ws);

    // 3) Z = Ys*Ar^T + Yd*Br^T   (18x4 tiles, one wave each)
    z_gemm_kernel<<<72, 32, 0, stream>>>(ws, ws);

    // 4) Me = Xsum * Z^T         (18x18 tiles)
    me_gemm_kernel<<<dim3(18, 18), 32, 0, stream>>>(ws, ws);

    // 5) Mp = U1 * U2^T          (5x5 tiles, guarded)
    mp_gemm_kernel<<<dim3(5, 5), 32, 0, stream>>>(U1, U2, ws);

    // 6) scatter edge-pair contributions (1.33M atomic f32 adds)
    scatter_kernel<<<(NE * NE * 4 + 255) / 256, 256, 0, stream>>>(ws, src1, dst1, src2, dst2, out);

    // 7) diagonal Mp add
    diag_kernel<<<(NM + 255) / 256, 256, 0, stream>>>(ws, out);
}